// SelfAttention_5239860101854
// MI455X (gfx1250) — compile-verified
//
#include <hip/hip_runtime.h>
#include <hip/hip_bf16.h>

// Problem constants (from reference setup_inputs)
#define B_   4
#define T_   16
#define C_   64
#define CQ_  8
#define HW_  9216          // 96*96
#define PG_  16            // pixels per group (WMMA N)
#define NG_  (HW_ / PG_)   // 576 groups per batch
#define KQK_ 192           // 3 taps * 64 channels

typedef float v2f __attribute__((ext_vector_type(2)));
typedef float v8f __attribute__((ext_vector_type(8)));
typedef uint32_t u32x4 __attribute__((ext_vector_type(4)));
typedef uint32_t u32x8 __attribute__((ext_vector_type(8)));

// D = A(16x4) * B(4x16) + C(16x16), all f32.
__device__ __forceinline__ v8f wmma4(v2f a, v2f b, v8f c) {
    return __builtin_amdgcn_wmma_f32_16x16x4_f32(false, a, false, b, (short)0, c,
                                                 false, false);
}

// ---------------------------------------------------------------------------
// Tensor Data Mover: DMA a 2D tile (1024 rows x 16 f32, row stride HW_*4 B)
// from global memory into LDS.  D# per CDNA5 ISA 8.3/8.4.  Issue once per WG
// (gated to wave 0); pair with tdm_wait() before the LDS tile is consumed.
// ---------------------------------------------------------------------------
__device__ __forceinline__ void tdm_load_tile(uint32_t lds_addr,
                                              const float* gptr)
{
    const uint64_t ga = (uint64_t)(uintptr_t)gptr;
    u32x4 g0;
    g0.x = 1u;                                   // count=1, user descriptor
    g0.y = lds_addr;                             // LDS byte address
    g0.z = (uint32_t)ga;                         // global_addr[31:0]
    g0.w = (uint32_t)(ga >> 32) | (2u << 30);    // global_addr[56:32] | type=2
    u32x8 g1;
    g1.s0 = (2u << 16);                          // wg_mask=0, data_size=4B
    g1.s1 = ((uint32_t)PG_) << 16;               // tensor_dim0 = 16
    g1.s2 = ((uint32_t)(T_ * C_) & 0xFFFFu) << 16; // tensor_dim1 = 1024 (lo16)
    g1.s3 = ((uint32_t)PG_) << 16;               // tensor_dim1 hi16=0 | tile_dim0=16
    g1.s4 = (uint32_t)(T_ * C_);                 // tile_dim1 = 1024, tile_dim2 = 0
    g1.s5 = (uint32_t)HW_;                       // tensor_dim0_stride lo32 = 9216
    g1.s6 = 0u;                                  // stride hi | dim1_stride lo
    g1.s7 = 0u;
    asm volatile("tensor_load_to_lds %0, %1" :: "s"(g0), "s"(g1) : "memory");
}

__device__ __forceinline__ void tdm_wait() {
    asm volatile("s_wait_tensorcnt 0x0" ::: "memory");
}

// ---------------------------------------------------------------------------
// Kernel 1: temporal conv -> Q,K (packed as 16 "channels") and partial energy.
// One WG = one (batch, 16-pixel group). 256 threads = 8 waves.
// LDS: A[16][192] (12KB) | Xt[18 slots][64c][16p] (72KB, slots 0/17 zero pad)
//      | QKs[16t][16m][16p] (16KB) | Elds[8][256] (8KB)  => 108KB dynamic LDS.
// ---------------------------------------------------------------------------
__global__ __launch_bounds__(256)
void qk_energy_kernel(const float* __restrict__ x,
                      const float* __restrict__ wq, const float* __restrict__ bq,
                      const float* __restrict__ wk, const float* __restrict__ bk,
                      float* __restrict__ e_part)
{
    extern __shared__ float smem[];
    float* Ald  = smem;                          // 16*192
    float* Xt   = Ald + 16 * KQK_;               // (T_+2)*64*16, slot = t+1
    float* QKs  = Xt + (T_ + 2) * C_ * PG_;      // 16*16*16
    float* Elds = QKs + T_ * 16 * PG_;           // 8*256

    const int tid  = threadIdx.x;
    const int lane = tid & 31;
    const int wave = tid >> 5;
    const int row  = lane & 15;
    const int hi   = lane >> 4;
    const int b    = blockIdx.y;
    const int p0   = blockIdx.x * PG_;

    // Async DMA of the 64KB X tile into slots 1..16 (overlaps staging below).
    const float* xb = x + (size_t)b * T_ * C_ * HW_ + p0;
    if (wave == 0) {
        tdm_load_tile((uint32_t)(uintptr_t)(Xt + C_ * PG_), xb);
    }
    // Zero the temporal halo slots (t = -1 and t = 16).
    for (int i = tid; i < C_ * PG_; i += 256) {
        Xt[i] = 0.f;
        Xt[(T_ + 1) * C_ * PG_ + i] = 0.f;
    }
    // Stage A = [Wq; Wk], A[m][kd*64+c]
    for (int i = tid; i < 16 * KQK_; i += 256) {
        int m  = i / KQK_;
        int k  = i % KQK_;
        int kd = k >> 6;
        int c  = k & 63;
        Ald[i] = (m < 8) ? wq[(m * C_ + c) * 3 + kd]
                         : wk[((m - 8) * C_ + c) * 3 + kd];
    }
    // Per-lane channel biases for the C-fragment rows this lane owns.
    float biasv[8];
#pragma unroll
    for (int r = 0; r < 8; ++r) biasv[r] = hi ? bk[r] : bq[r];

    if (wave == 0) tdm_wait();
    __syncthreads();

    // Conv phase: each wave computes 2 time steps. QK(16m x 16p) = A @ Xpatch.
    // Branch-free: halo slots supply the conv zero padding.
    for (int ti = 0; ti < 2; ++ti) {
        int t = wave * 2 + ti;
        v8f acc = {};
        for (int ks = 0; ks < KQK_ / 4; ++ks) {
            int k  = ks * 4 + hi * 2;
            int kd = k >> 6;             // pair never crosses a tap boundary
            int c  = k & 63;
            v2f a, bf;
            a.x  = Ald[row * KQK_ + k];
            a.y  = Ald[row * KQK_ + k + 1];
            const float* xr = &Xt[((t + kd) * C_ + c) * PG_ + row];
            bf.x = xr[0];
            bf.y = xr[PG_];              // channel c+1, 64B apart
            acc = wmma4(a, bf, acc);
        }
        // C-frag: VGPR r -> M = hi*8+r (channel m), N = row (pixel)
#pragma unroll
        for (int r = 0; r < 8; ++r) {
            int m = hi * 8 + r;
            QKs[(t * 16 + m) * PG_ + row] = acc[r] + biasv[r];
        }
    }
    __syncthreads();

    // Energy phase: E[t,s] = sum_{m<8,p} Q[t,m,p]*K[s,m,p].
    // K-dim flat kappa = m*16+p (128 values) -> 32 wmma split over 8 waves.
    v8f eacc = {};
#pragma unroll
    for (int i = 0; i < 4; ++i) {
        int kap = (wave * 4 + i) * 4 + hi * 2;
        v2f a, bf;
        a.x  = QKs[row * 256 + kap];          // Q[t=row][kappa]
        a.y  = QKs[row * 256 + kap + 1];
        bf.x = QKs[row * 256 + 128 + kap];    // K[s=row][kappa]
        bf.y = QKs[row * 256 + 128 + kap + 1];
        eacc = wmma4(a, bf, eacc);
    }
#pragma unroll
    for (int r = 0; r < 8; ++r) {
        int M = hi * 8 + r;
        Elds[wave * 256 + M * 16 + row] = eacc[r];
    }
    __syncthreads();

    // Fixed-order cross-wave reduction (deterministic; no FP atomics).
    float s = 0.f;
#pragma unroll
    for (int w = 0; w < 8; ++w) s += Elds[w * 256 + tid];
    e_part[((size_t)(b * NG_ + blockIdx.x)) * 256 + tid] = s;
}

// ---------------------------------------------------------------------------
// Kernel 2: reduce 576 partials per batch, softmax over s. Grid = 4 blocks.
// ---------------------------------------------------------------------------
__global__ __launch_bounds__(256)
void softmax_kernel(const float* __restrict__ e_part, float* __restrict__ heat)
{
    __shared__ float E[256];
    const int b   = blockIdx.x;
    const int tid = threadIdx.x;      // tid = t*16 + s
    const float* ep = e_part + (size_t)b * NG_ * 256;
    float s = 0.f;
    for (int g = 0; g < NG_; ++g) s += ep[g * 256 + tid];
    E[tid] = s;
    __syncthreads();
    const int t = tid >> 4;
    float m = -1e30f;
#pragma unroll
    for (int j = 0; j < 16; ++j) m = fmaxf(m, E[t * 16 + j]);
    float den = 0.f;
#pragma unroll
    for (int j = 0; j < 16; ++j) den += __expf(E[t * 16 + j] - m);
    heat[b * 256 + tid] = __expf(E[tid] - m) / den;
}

// ---------------------------------------------------------------------------
// Kernel 3: out = gamma*(Wv @ (heat @ X) + bv) + x  (per pixel group).
// LDS: Hs(1KB) | Bvs(256B) | Wvs(16KB) | Xt(64KB) | Y(64KB) => ~145.5KB.
// ---------------------------------------------------------------------------
__global__ __launch_bounds__(256)
void av_out_kernel(const float* __restrict__ x,
                   const float* __restrict__ wv, const float* __restrict__ bv,
                   const float* __restrict__ gamma,
                   const float* __restrict__ heat,
                   float* __restrict__ out)
{
    extern __shared__ float smem[];
    float* Hs  = smem;                 // 256
    float* Bvs = Hs + 256;             // 64
    float* Wvs = Bvs + 64;             // 64*64
    float* Xt  = Wvs + 64 * 64;        // 16*64*16  [s][c][p]
    float* Y   = Xt + T_ * C_ * PG_;   // 64*16*16  [c2][t][p]

    const int tid  = threadIdx.x;
    const int lane = tid & 31;
    const int wave = tid >> 5;
    const int row  = lane & 15;
    const int hi   = lane >> 4;
    const int b    = blockIdx.y;
    const int p0   = blockIdx.x * PG_;

    // Async DMA of the 64KB X tile; overlaps weight/heat staging.
    const float* xb = x + (size_t)b * T_ * C_ * HW_ + p0;
    if (wave == 0) {
        tdm_load_tile((uint32_t)(uintptr_t)Xt, xb);
    }
    Hs[tid] = heat[b * 256 + tid];
    if (tid < 64) Bvs[tid] = bv[tid];
    for (int i = tid; i < 64 * 64; i += 256) Wvs[i] = wv[i];
    if (wave == 0) tdm_wait();
    __syncthreads();

    // Step A: Y[c2](16t x 16p) = heat(16x16) @ X[c2](16s x 16p); 8 c2 per wave
    for (int c2 = wave; c2 < C_; c2 += 8) {
        v8f acc = {};
#pragma unroll
        for (int ks = 0; ks < 4; ++ks) {
            int s = ks * 4 + hi * 2;
            v2f a, bf;
            a.x  = Hs[row * 16 + s];
            a.y  = Hs[row * 16 + s + 1];
            bf.x = Xt[(s * C_ + c2) * PG_ + row];
            bf.y = Xt[((s + 1) * C_ + c2) * PG_ + row];
            acc = wmma4(a, bf, acc);
        }
#pragma unroll
        for (int r = 0; r < 8; ++r) {
            int t = hi * 8 + r;
            Y[(c2 * 16 + t) * PG_ + row] = acc[r];
        }
    }
    __syncthreads();

    const float g = gamma[0];

    // Step B: Out[c](t fixed, 16p) = Wv(64x64) @ Y(64 x 16p); 64 (mt,t) tiles
    for (int pair = wave; pair < 64; pair += 8) {
        int mt = pair >> 4;   // M-tile of c
        int t  = pair & 15;
        v8f acc = {};
        for (int ks = 0; ks < 16; ++ks) {
            int k = ks * 4 + hi * 2;   // c2
            v2f a, bf;
            a.x  = Wvs[(mt * 16 + row) * 64 + k];
            a.y  = Wvs[(mt * 16 + row) * 64 + k + 1];
            bf.x = Y[(k * 16 + t) * PG_ + row];
            bf.y = Y[((k + 1) * 16 + t) * PG_ + row];
            acc = wmma4(a, bf, acc);
        }
        // Fused epilogue: gamma*(acc + bv) + residual (x from LDS tile)
#pragma unroll
        for (int r = 0; r < 8; ++r) {
            int c = mt * 16 + hi * 8 + r;
            float val = g * (acc[r] + Bvs[c]) + Xt[(t * C_ + c) * PG_ + row];
            out[(size_t)((b * T_ + t) * C_ + c) * HW_ + p0 + row] = val;
        }
    }
}

// ---------------------------------------------------------------------------
extern "C" void kernel_launch(void* const* d_in, const int* in_sizes, int n_in,
                              void* d_out, int out_size, void* d_ws, size_t ws_size,
                              hipStream_t stream)
{
    const float* x     = (const float*)d_in[0];
    const float* wq    = (const float*)d_in[1];
    const float* bq    = (const float*)d_in[2];
    const float* wk    = (const float*)d_in[3];
    const float* bk    = (const float*)d_in[4];
    const float* wv    = (const float*)d_in[5];
    const float* bv    = (const float*)d_in[6];
    const float* gamma = (const float*)d_in[7];
    float* out = (float*)d_out;

    float* e_part = (float*)d_ws;                       // 4*576*256 f32 = 2.25MB
    float* heat   = e_part + (size_t)B_ * NG_ * 256;    // 1024 f32

    const size_t smem1 = (size_t)(16 * KQK_ + (T_ + 2) * C_ * PG_ +
                                  T_ * 16 * PG_ + 8 * 256) * 4;
    const size_t smem3 = (size_t)(256 + 64 + 64 * 64 + T_ * C_ * PG_ +
                                  C_ * T_ * PG_) * 4;

    dim3 grid(NG_, B_);
    qk_energy_kernel<<<grid, 256, smem1, stream>>>(x, wq, bq, wk, bk, e_part);
    softmax_kernel<<<B_, 256, 0, stream>>>(e_part, heat);
    av_out_kernel<<<grid, 256, smem3, stream>>>(x, wv, bv, gamma, heat, out);
}